// SJLTProjection_12584254177373
// MI455X (gfx1250) — compile-verified
//
#include <hip/hip_runtime.h>
#include <cstdint>

// Problem constants (from reference)
#define B_  64
#define D_  262144
#define C_  4
#define P_  4096

// Tiling
#define BG        8                      // batches privatized per workgroup (128 KB LDS acc)
#define THREADS   256                    // 8 wave32s
#define CHUNKS    32                     // D split into 32 chunks of 8192
#define CHUNK_LEN (D_ / CHUNKS)          // 8192
#define NTILES    (CHUNK_LEN / THREADS)  // 32 tiles of 256 d's

// ---------------------------------------------------------------------------
// Scatter kernel (emitted first so the disasm snippet shows the hot loop).
// grid = (CHUNKS, B/BG). Each block owns a D-chunk for a group of BG batches,
// accumulating into LDS with ds_add_f32, then flushing with
// global_atomic_add_f32. idx/sign tiles are double-buffered into LDS with the
// CDNA5 async copy engine (global_load_async_to_lds_b128 + s_wait_asynccnt).
// ---------------------------------------------------------------------------
__global__ __launch_bounds__(THREADS) void sjlt_scatter(
    const float* __restrict__ x,          // [B, D] f32
    const long long* __restrict__ idx,    // [D, C] i64, values in [0, P)
    const int* __restrict__ sgn,          // [D, C] i32, values in {0,1}
    float* __restrict__ out)              // [B, P] f32 (pre-zeroed)
{
  __shared__ float       s_acc[BG * P_];          // 128 KB accumulators
  __shared__ ulonglong2  s_idx[2][THREADS * 2];   // 16 KB staged indices (32 B/d)
  __shared__ uint4       s_sgn[2][THREADS];       //  8 KB staged signs   (16 B/d)

  const int tid       = threadIdx.x;
  const int chunkBase = blockIdx.x * CHUNK_LEN;
  const int bgBase    = blockIdx.y * BG;

  // ---- zero LDS accumulators (vectorized: 32 float4 stores per thread) ----
  {
    float4* accv = reinterpret_cast<float4*>(s_acc);
    for (int i = tid; i < (BG * P_) / 4; i += THREADS)
      accv[i] = make_float4(0.f, 0.f, 0.f, 0.f);
  }
  __syncthreads();

  const unsigned long long idxBase = (unsigned long long)idx;
  const unsigned long long sgnBase = (unsigned long long)sgn;

  // Stage one tile (256 d's) of idx+sign into LDS buffer `buf` using the
  // async copy engine (ASYNCcnt-tracked, no VGPR data path). Each lane copies
  // the 32B of indices and 16B of signs it will itself consume, so a per-wave
  // s_wait_asynccnt is sufficient -- no barrier needed.
  auto stage = [&](int t, int buf) {
    const int d = chunkBase + t * THREADS + tid;
    const uint32_t gi = (uint32_t)d * 32u;   // byte offset into idx (C*8B per d)
    const uint32_t gs = (uint32_t)d * 16u;   // byte offset into sgn (C*4B per d)
    const uint32_t li = (uint32_t)(uintptr_t)(&s_idx[buf][tid * 2]);
    const uint32_t ls = (uint32_t)(uintptr_t)(&s_sgn[buf][tid]);
    asm volatile("global_load_async_to_lds_b128 %0, %1, %2"
                 :: "v"(li), "v"(gi), "s"(idxBase) : "memory");
    asm volatile("global_load_async_to_lds_b128 %0, %1, %2"
                 :: "v"(li + 16u), "v"(gi + 16u), "s"(idxBase) : "memory");
    asm volatile("global_load_async_to_lds_b128 %0, %1, %2"
                 :: "v"(ls), "v"(gs), "s"(sgnBase) : "memory");
  };

  // Prologue: stage tile 0.
  stage(0, 0);

  for (int t = 0; t < NTILES; ++t) {
    // Issue the next tile, then wait for the current tile's 3 async ops
    // (async loads complete in order, so waiting for <=3 outstanding means
    // the oldest 3 -- the current tile -- have landed in LDS).
    if (t + 1 < NTILES) {
      stage(t + 1, (t + 1) & 1);
      asm volatile("s_wait_asynccnt 0x3" ::: "memory");
    } else {
      asm volatile("s_wait_asynccnt 0x0" ::: "memory");
    }

    const int buf = t & 1;
    const ulonglong2 i01 = s_idx[buf][tid * 2];
    const ulonglong2 i23 = s_idx[buf][tid * 2 + 1];
    const uint4      sg  = s_sgn[buf][tid];

    const uint32_t p0 = (uint32_t)i01.x & (P_ - 1);
    const uint32_t p1 = (uint32_t)i01.y & (P_ - 1);
    const uint32_t p2 = (uint32_t)i23.x & (P_ - 1);
    const uint32_t p3 = (uint32_t)i23.y & (P_ - 1);

    const int d = chunkBase + t * THREADS + tid;
    const float* xp = x + (size_t)bgBase * D_ + d;

    // 8 batches * 4 hash positions -> 32 ds_add_f32 LDS atomics per d.
    // sgn ? xv : -xv lowers to a single v_cndmask with a source-neg modifier.
#pragma unroll
    for (int b = 0; b < BG; ++b) {
      const float xv = xp[(size_t)b * D_];     // coalesced global_load_b32
      float* a = s_acc + (b << 12);            // b * P_
      atomicAdd(a + p0, sg.x ? xv : -xv);
      atomicAdd(a + p1, sg.y ? xv : -xv);
      atomicAdd(a + p2, sg.z ? xv : -xv);
      atomicAdd(a + p3, sg.w ? xv : -xv);
    }
  }

  __syncthreads();

  // ---- flush: LDS partials -> global (L2 atomic fadd), coalesced over p ----
  float* ob = out + (size_t)bgBase * P_;
  for (int i = tid; i < BG * P_; i += THREADS)
    unsafeAtomicAdd(ob + i, s_acc[i]);
}

// ---------------------------------------------------------------------------
// Zero the output (harness poisons d_out with 0xAA before timing).
// 256 blocks * 256 threads * float4 = 262144 floats = B*P
// ---------------------------------------------------------------------------
__global__ __launch_bounds__(THREADS) void sjlt_zero(float* __restrict__ out) {
  const int i = blockIdx.x * THREADS + threadIdx.x;
  reinterpret_cast<float4*>(out)[i] = make_float4(0.f, 0.f, 0.f, 0.f);
}

// ---------------------------------------------------------------------------
// Launcher
// ---------------------------------------------------------------------------
extern "C" void kernel_launch(void* const* d_in, const int* in_sizes, int n_in,
                              void* d_out, int out_size, void* d_ws, size_t ws_size,
                              hipStream_t stream) {
  (void)in_sizes; (void)n_in; (void)out_size; (void)d_ws; (void)ws_size;

  const float*     x   = (const float*)d_in[0];      // [B, D] f32
  const long long* idx = (const long long*)d_in[1];  // [D, C] i64
  const int*       sgn = (const int*)d_in[2];        // [D, C] i32
  float*           out = (float*)d_out;              // [B, P] f32

  // Zero output: (B*P)/(THREADS*4) = 256 blocks of 256 threads, float4 each.
  sjlt_zero<<<dim3((B_ * P_) / (THREADS * 4)), dim3(THREADS), 0, stream>>>(out);

  // Scatter: 32 D-chunks x 8 batch-groups.
  sjlt_scatter<<<dim3(CHUNKS, B_ / BG), dim3(THREADS), 0, stream>>>(x, idx, sgn, out);
}